// HashGrid_438086664221
// MI455X (gfx1250) — compile-verified
//
#include <hip/hip_runtime.h>
#include <stdint.h>
#include <stddef.h>

typedef float v2f __attribute__((ext_vector_type(2)));
// gcc-style vector to match the builtin's parameter type exactly
// ('__attribute__((__vector_size__(2 * sizeof(int)))) int' per the diagnostic)
typedef int gv2i __attribute__((vector_size(2 * sizeof(int))));
typedef __attribute__((address_space(1))) gv2i gv2i_global;
typedef __attribute__((address_space(3))) gv2i gv2i_lds;

// ---- Compile-time replica of the reference config ----------------------
// b = 32^(1/15) = 2^(1/3);  LODS[l] = 1 + floor(16 * 2^(l/3))
constexpr int      kNumLod        = 16;
constexpr int      kRes[kNumLod]  = {17, 21, 26, 33, 41, 51, 65, 81,
                                     102, 129, 162, 204, 257, 323, 407, 513};
// table size = min(res^3, 2^19)
constexpr unsigned kSize[kNumLod] = {4913u, 9261u, 17576u, 35937u, 68921u,
                                     132651u, 274625u,
                                     524288u, 524288u, 524288u, 524288u,
                                     524288u, 524288u, 524288u, 524288u, 524288u};

// LODs 0..2 staged into LDS: 4913+9261+17576 = 31750 float2 entries = 254000 B
constexpr int      kLdsLods              = 3;
constexpr unsigned kLdsBase[kLdsLods]    = {0u, 4913u, 14174u};
constexpr unsigned kLdsEntries           = 31750u;
constexpr size_t   kLdsBytes             = (size_t)kLdsEntries * sizeof(v2f);

struct Tables { const float* t[kNumLod]; };

// ---- gfx1250 async global->LDS staging (guarded) -----------------------
__device__ __forceinline__ void stage_entry(const v2f* __restrict__ g, v2f* l) {
#if __has_builtin(__builtin_amdgcn_global_load_async_to_lds_b64)
  __builtin_amdgcn_global_load_async_to_lds_b64(
      (gv2i_global*)g,       // addrspace(1) int2* per compiler diagnostic
      (gv2i_lds*)l,          // addrspace(3) side
      /*offset=*/0, /*cpol=*/0);
#else
  *l = *g;  // fallback: global_load_b64 + ds_store_b64
#endif
}

__device__ __forceinline__ void wait_staging_done() {
#if __has_builtin(__builtin_amdgcn_global_load_async_to_lds_b64)
#if __has_builtin(__builtin_amdgcn_s_wait_asynccnt)
  __builtin_amdgcn_s_wait_asynccnt(0);
#else
  asm volatile("s_wait_asynccnt 0" ::: "memory");
#endif
#endif
}

// ---- One LOD: 8 hashes, 8 gathers, trilinear combine -------------------
template <int L>
__device__ __forceinline__ v2f lod_interp(const Tables& tabs, const float* s_tab,
                                          float px, float py, float pz) {
  constexpr int      RES = kRes[L];
  constexpr unsigned SZ  = kSize[L];
  constexpr float    S   = (float)(RES - 1);

  const v2f* tab;
  if constexpr (L < kLdsLods)
    tab = (const v2f*)s_tab + kLdsBase[L];        // LDS-resident table
  else
    tab = (const v2f*)tabs.t[L];                  // L2-resident table

  const float gx = px * S, gy = py * S, gz = pz * S;
  const float fx = floorf(gx), fy = floorf(gy), fz = floorf(gz);
  const int   x0 = (int)fx, y0 = (int)fy, z0 = (int)fz;
  const float tx = gx - fx, ty = gy - fy, tz = gz - fz;
  const int   x1 = min(x0 + 1, RES - 1);
  const int   y1 = min(y0 + 1, RES - 1);
  const int   z1 = min(z0 + 1, RES - 1);

  // Instant-NGP hash: (x*1) ^ (y*2654435761) ^ (z*805459861), then % SZ
  const unsigned hx0 = (unsigned)x0;
  const unsigned hx1 = (unsigned)x1;
  const unsigned hy0 = (unsigned)y0 * 2654435761u;
  const unsigned hy1 = (unsigned)y1 * 2654435761u;
  const unsigned hz0 = (unsigned)z0 * 805459861u;
  const unsigned hz1 = (unsigned)z1 * 805459861u;

  // Issue all 8 gathers before consuming (lets the compiler pipeline loads).
  const v2f f000 = tab[(hx0 ^ hy0 ^ hz0) % SZ];
  const v2f f001 = tab[(hx0 ^ hy0 ^ hz1) % SZ];
  const v2f f010 = tab[(hx0 ^ hy1 ^ hz0) % SZ];
  const v2f f011 = tab[(hx0 ^ hy1 ^ hz1) % SZ];
  const v2f f100 = tab[(hx1 ^ hy0 ^ hz0) % SZ];
  const v2f f101 = tab[(hx1 ^ hy0 ^ hz1) % SZ];
  const v2f f110 = tab[(hx1 ^ hy1 ^ hz0) % SZ];
  const v2f f111 = tab[(hx1 ^ hy1 ^ hz1) % SZ];

  const float ux = 1.0f - tx, uy = 1.0f - ty, uz = 1.0f - tz;
  v2f acc = f000 * (ux * uy * uz);
  acc += f001 * (ux * uy * tz);
  acc += f010 * (ux * ty * uz);
  acc += f011 * (ux * ty * tz);
  acc += f100 * (tx * uy * uz);
  acc += f101 * (tx * uy * tz);
  acc += f110 * (tx * ty * uz);
  acc += f111 * (tx * ty * tz);
  return acc;
}

// Unrolled over all 16 LODs; store each result immediately (NT) to keep
// VGPR live ranges short.
template <int L>
__device__ __forceinline__ void do_all_lods(const Tables& tabs, const float* s_tab,
                                            float px, float py, float pz, v2f* orow) {
  v2f r = lod_interp<L>(tabs, s_tab, px, py, pz);
  __builtin_nontemporal_store(r, orow + L);   // don't let 128 MB output evict tables
  if constexpr (L + 1 < kNumLod)
    do_all_lods<L + 1>(tabs, s_tab, px, py, pz, orow);
}

__global__ __launch_bounds__(1024, 1) void HashGrid_438086664221_kernel(
    const float* __restrict__ pts, Tables tabs, float* __restrict__ out, int n) {
  extern __shared__ float s_tab[];
  const int tid = threadIdx.x;

  // --- Stage the 3 smallest tables into LDS (async DMA path) ---
  {
    const v2f* g0 = (const v2f*)tabs.t[0];
    const v2f* g1 = (const v2f*)tabs.t[1];
    const v2f* g2 = (const v2f*)tabs.t[2];
    v2f* s = (v2f*)s_tab;
    for (unsigned i = tid; i < kSize[0]; i += blockDim.x)
      stage_entry(g0 + i, s + kLdsBase[0] + i);
    for (unsigned i = tid; i < kSize[1]; i += blockDim.x)
      stage_entry(g1 + i, s + kLdsBase[1] + i);
    for (unsigned i = tid; i < kSize[2]; i += blockDim.x)
      stage_entry(g2 + i, s + kLdsBase[2] + i);
    wait_staging_done();
    __syncthreads();
  }

  // --- Persistent grid-stride loop over points ---
  const int stride = (int)(gridDim.x * blockDim.x);
  for (int p = (int)(blockIdx.x * blockDim.x) + tid; p < n; p += stride) {
    const float* pp = pts + 3u * (size_t)p;
    // points are streamed exactly once: non-temporal
    const float px = __builtin_nontemporal_load(pp + 0);
    const float py = __builtin_nontemporal_load(pp + 1);
    const float pz = __builtin_nontemporal_load(pp + 2);
    v2f* orow = (v2f*)(out + 32u * (size_t)p);
    do_all_lods<0>(tabs, s_tab, px, py, pz, orow);
  }
}

extern "C" void kernel_launch(void* const* d_in, const int* in_sizes, int n_in,
                              void* d_out, int out_size, void* d_ws, size_t ws_size,
                              hipStream_t stream) {
  (void)d_ws; (void)ws_size; (void)out_size;
  if (n_in < 1 + kNumLod) return;

  const float* pts = (const float*)d_in[0];
  Tables tabs;
  for (int l = 0; l < kNumLod; ++l) tabs.t[l] = (const float*)d_in[1 + l];
  float* out = (float*)d_out;
  const int n = in_sizes[0] / 3;

  // 254 KB dynamic LDS (> default 64 KB cap): opt in. Capture-safe (not a
  // stream op), deterministic, idempotent.
  hipError_t err = hipFuncSetAttribute(
      reinterpret_cast<const void*>(&HashGrid_438086664221_kernel),
      hipFuncAttributeMaxDynamicSharedMemorySize, (int)kLdsBytes);
  (void)err;

  // 1024 threads = 32 wave32 = 8 waves/SIMD; one block/WGP (LDS-limited).
  // 256 blocks -> ~4 points/thread grid-stride, amortizing LDS staging.
  const int blocks = 256;
  HashGrid_438086664221_kernel<<<blocks, 1024, kLdsBytes, stream>>>(pts, tabs, out, n);
}